// GraphAttentionAttr_73701638799909
// MI455X (gfx1250) — compile-verified
//
#include <hip/hip_runtime.h>
#include <hip/hip_bf16.h>

// Shapes (fixed by the reference): B=64, O=100, A=16, D=1024, H=512
#define BO   6400      // B*O
#define DIM  1024      // D
#define HID  512       // H
#define NATT 16        // A
#define EPSF 1e-8f

typedef __attribute__((ext_vector_type(2))) float v2f;
typedef __attribute__((ext_vector_type(8))) float v8f;

// Fast tanh: prefer the CDNA5 hardware TRANS op (v_tanh_f32) when clang
// exposes it; otherwise an overflow-safe exp-based form using the hardware
// v_exp_f32 (t in (0,1], so no inf/NaN for large |x|).
__device__ __forceinline__ float fast_tanh(float x) {
#if __has_builtin(__builtin_amdgcn_tanhf)
    return __builtin_amdgcn_tanhf(x);
#elif __has_builtin(__builtin_amdgcn_tanh_f32)
    return __builtin_amdgcn_tanh_f32(x);
#else
    float t = __expf(-2.0f * __builtin_fabsf(x));
    float r = (1.0f - t) / (1.0f + t);
    return __builtin_copysignf(r, x);
#endif
}

// -----------------------------------------------------------------------------
// Kernel 1: att_obj[BO,H] = obj[BO,D] * W_obj[H,D]^T + b_obj   (f32 WMMA)
//
// V_WMMA_F32_16X16X4_F32 fragment layout (wave32):
//   A (16x4):  lane L<16 holds A[M=L][K=0,1] in {x,y}; lane L+16 holds K=2,3
//   B (4x16):  lane L<16 holds B[K=0,1][N=L];          lane L+16 holds K=2,3
//   C/D:       VGPR r, lanes 0-15 -> D[M=r][N=lane]; lanes 16-31 -> D[M=r+8]
// B[k][n] = W_obj[n][k]  (W_obj row-major [H,D] is column-major [D,H]),
// so both A and B fragments are contiguous float2 loads per lane.
// One wave computes a 16x64 tile (A fragment reused across 4 WMMAs).
// -----------------------------------------------------------------------------
__global__ __launch_bounds__(256) void gemm_att_obj(
    const float* __restrict__ obj,    // [BO, D]
    const float* __restrict__ Wobj,   // [H, D]
    const float* __restrict__ bobj,   // [H]
    float* __restrict__ att)          // [BO, H]
{
    const int wave = threadIdx.x >> 5;
    const int lane = threadIdx.x & 31;
    const int gw   = blockIdx.x * 8 + wave;   // 0 .. (BO/16)*(H/64) - 1 = 3199
    const int m0   = (gw >> 3) << 4;          // row tile * 16
    const int n0   = (gw & 7) << 6;           // col tile * 64
    const int l15  = lane & 15;
    const int hi   = lane >> 4;               // 0: K half 0-1, 1: K half 2-3

    const float* aptr  = obj  + (size_t)(m0 + l15)        * DIM + 2 * hi;
    const float* bptr0 = Wobj + (size_t)(n0 +  0 + l15)   * DIM + 2 * hi;
    const float* bptr1 = Wobj + (size_t)(n0 + 16 + l15)   * DIM + 2 * hi;
    const float* bptr2 = Wobj + (size_t)(n0 + 32 + l15)   * DIM + 2 * hi;
    const float* bptr3 = Wobj + (size_t)(n0 + 48 + l15)   * DIM + 2 * hi;

    v8f c0 = {}, c1 = {}, c2 = {}, c3 = {};

    for (int k = 0; k < DIM; k += 4) {
        v2f a  = *(const v2f*)(aptr  + k);
        v2f b0 = *(const v2f*)(bptr0 + k);
        v2f b1 = *(const v2f*)(bptr1 + k);
        v2f b2 = *(const v2f*)(bptr2 + k);
        v2f b3 = *(const v2f*)(bptr3 + k);
        c0 = __builtin_amdgcn_wmma_f32_16x16x4_f32(false, a, false, b0, (short)0, c0, false, false);
        c1 = __builtin_amdgcn_wmma_f32_16x16x4_f32(false, a, false, b1, (short)0, c1, false, false);
        c2 = __builtin_amdgcn_wmma_f32_16x16x4_f32(false, a, false, b2, (short)0, c2, false, false);
        c3 = __builtin_amdgcn_wmma_f32_16x16x4_f32(false, a, false, b3, (short)0, c3, false, false);
    }

    const float bb0 = bobj[n0 +  0 + l15];
    const float bb1 = bobj[n0 + 16 + l15];
    const float bb2 = bobj[n0 + 32 + l15];
    const float bb3 = bobj[n0 + 48 + l15];

    const int rowbase = m0 + hi * 8;
#pragma unroll
    for (int r = 0; r < 8; ++r) {
        const size_t off = (size_t)(rowbase + r) * HID;
        att[off + n0 +  0 + l15] = c0[r] + bb0;
        att[off + n0 + 16 + l15] = c1[r] + bb1;
        att[off + n0 + 32 + l15] = c2[r] + bb2;
        att[off + n0 + 48 + l15] = c3[r] + bb3;
    }
}

// -----------------------------------------------------------------------------
// Kernel 2: fused tanh-score + softmax + masked renorm + weighted pooling.
// One 256-thread block per (b,o). p_attr_feats and attr_feats are each read
// exactly once (this kernel sets the HBM-bandwidth floor: ~630 MB streamed,
// ~27 us at 23.3 TB/s).
// -----------------------------------------------------------------------------
__global__ __launch_bounds__(256) void attn_pool(
    const float* __restrict__ att,     // [BO, H]   (from kernel 1)
    const float* __restrict__ p,       // [BO, A, H]
    const float* __restrict__ feats,   // [BO, A, D]
    const int*   __restrict__ masks,   // [BO, A]
    const float* __restrict__ walpha,  // [H]
    float* __restrict__ out)           // [BO, D]
{
    __shared__ float s_att[HID];
    __shared__ float s_wa[HID];
    __shared__ float s_sc[NATT];
    __shared__ float s_w[NATT];

    const int bo = blockIdx.x;
    const int t  = threadIdx.x;

    // Stage 0: att_obj row + w_alpha into LDS
    s_att[t]       = att[(size_t)bo * HID + t];
    s_att[t + 256] = att[(size_t)bo * HID + t + 256];
    s_wa[t]        = walpha[t];
    s_wa[t + 256]  = walpha[t + 256];
    __syncthreads();

    // Stage 1: scores[a] = sum_h tanh(p + att) * w_alpha
    // wave w: lanes 0-15 -> attr 2w, lanes 16-31 -> attr 2w+1
    const int wave = t >> 5;
    const int lane = t & 31;
    const int a    = wave * 2 + (lane >> 4);
    const int g    = lane & 15;
    const float* prow = p + ((size_t)bo * NATT + a) * HID;

    float acc = 0.0f;
#pragma unroll 4
    for (int h0 = g * 4; h0 < HID; h0 += 64) {
        float4 pv = *(const float4*)(prow + h0);
        float4 av = *(const float4*)(s_att + h0);
        float4 wv = *(const float4*)(s_wa + h0);
        acc += fast_tanh(pv.x + av.x) * wv.x;
        acc += fast_tanh(pv.y + av.y) * wv.y;
        acc += fast_tanh(pv.z + av.z) * wv.z;
        acc += fast_tanh(pv.w + av.w) * wv.w;
    }
    // reduce within each 16-lane group (xor masks <= 8 never cross the half)
    acc += __shfl_xor(acc, 8, 32);
    acc += __shfl_xor(acc, 4, 32);
    acc += __shfl_xor(acc, 2, 32);
    acc += __shfl_xor(acc, 1, 32);
    if (g == 0) s_sc[a] = acc;
    __syncthreads();

    // Stage 2: 16-lane softmax + mask + renorm (b_alpha cancels in softmax)
    if (t < NATT) {
        float s = s_sc[t];
        float m = s;
        m = fmaxf(m, __shfl_xor(m, 8, 32));
        m = fmaxf(m, __shfl_xor(m, 4, 32));
        m = fmaxf(m, __shfl_xor(m, 2, 32));
        m = fmaxf(m, __shfl_xor(m, 1, 32));
        float e = __expf(s - m);
        float sum = e;
        sum += __shfl_xor(sum, 8, 32);
        sum += __shfl_xor(sum, 4, 32);
        sum += __shfl_xor(sum, 2, 32);
        sum += __shfl_xor(sum, 1, 32);
        float w = e / sum;
        w *= (float)masks[bo * NATT + t];
        float ms = w;
        ms += __shfl_xor(ms, 8, 32);
        ms += __shfl_xor(ms, 4, 32);
        ms += __shfl_xor(ms, 2, 32);
        ms += __shfl_xor(ms, 1, 32);
        s_w[t] = w / (ms + EPSF);
    }
    __syncthreads();

    // Stage 3: out[d] = sum_a w[a] * feats[a][d]; thread t owns d = 4t..4t+3
    const float* frow = feats + (size_t)bo * NATT * DIM;
    const int d = t * 4;
    float4 r = make_float4(0.f, 0.f, 0.f, 0.f);
#pragma unroll
    for (int aa = 0; aa < NATT; ++aa) {
        const float w = s_w[aa];
        float4 f = *(const float4*)(frow + aa * DIM + d);
        r.x += w * f.x; r.y += w * f.y; r.z += w * f.z; r.w += w * f.w;
    }
    *(float4*)(out + (size_t)bo * DIM + d) = r;
}

extern "C" void kernel_launch(void* const* d_in, const int* in_sizes, int n_in,
                              void* d_out, int out_size, void* d_ws, size_t ws_size,
                              hipStream_t stream) {
    const float* obj    = (const float*)d_in[0];  // [B,O,D]
    const float* feats  = (const float*)d_in[1];  // [B,O,A,D]
    const float* pfeat  = (const float*)d_in[2];  // [B,O,A,H]
    const int*   masks  = (const int*)  d_in[3];  // [B,O,A]
    const float* Wobj   = (const float*)d_in[4];  // [H,D]
    const float* bobj   = (const float*)d_in[5];  // [H]
    const float* walpha = (const float*)d_in[6];  // [H]
    // d_in[7] = b_alpha: mathematically cancels inside softmax -> unused.

    float* att = (float*)d_ws;                    // BO*H floats = 13.1 MB scratch

    // (BO/16) * (H/64) = 400 * 8 = 3200 wave-tiles, 8 waves per block
    gemm_att_obj<<<400, 256, 0, stream>>>(obj, Wobj, bobj, att);
    attn_pool<<<BO, 256, 0, stream>>>(att, pfeat, feats, masks, walpha, (float*)d_out);
}